// FlexDecodingModel_20452634263887
// MI455X (gfx1250) — compile-verified
//
#include <hip/hip_runtime.h>
#include <math.h>

// ---------------- model constants (match reference) ----------------
#define D_MODEL 2048
#define N_HEADS 16
#define HEAD_DIM 128
#define SEQ 512
#define BATCH 2
#define TOKENS (BATCH * SEQ)   // 1024
#define FFN 8192
#define VOCAB 32000
#define LN_EPS 1e-5f
#define ATT_SCALE 0.08838834764831845f  // 128^-0.5

typedef __attribute__((ext_vector_type(2))) float v2f;
typedef __attribute__((ext_vector_type(8))) float v8f;

// CDNA5 async global->LDS copy path (ASYNCcnt-tracked), with sync fallback.
#if defined(__AMDGCN__) && __has_builtin(__builtin_amdgcn_global_load_async_to_lds_b128) && __has_builtin(__builtin_amdgcn_s_wait_asynccnt)
#define USE_ASYNC_LDS 1
typedef int v4i __attribute__((vector_size(16)));               // matches builtin proto
typedef __attribute__((address_space(1))) v4i* gas_v4i;         // global src
typedef __attribute__((address_space(3))) v4i* lds_v4i;         // LDS dst
#else
#define USE_ASYNC_LDS 0
#endif

#define AS_STRIDE 20            // A tile row stride (conflict-free, 16B-aligned rows)
#define BS_STRIDE 72            // B tile row stride (conflict-free)
#define AS_ELEMS (64 * AS_STRIDE)
#define BS_ELEMS (16 * BS_STRIDE)

// =====================================================================
// Generic fp32 WMMA GEMM:  C[M,N] = act(A[M,K] @ B[K,N] + bias) + residual
// act: 0 = none, 1 = exact GELU.  M,N multiples of 64; K multiple of 16.
// Block = 256 threads (8 waves); block tile 64x64; wave tile 16x32
// (two 16x16 f32 accumulators), V_WMMA_F32_16X16X4_F32 micro-steps.
// Grid: x = M/64 (fast dim -> weight-slab reuse in L2), y = N/64.
// Double-buffered LDS; async global->LDS copies when available.
// =====================================================================
__global__ __launch_bounds__(256)
void gemm_f32_wmma(const float* __restrict__ A, const float* __restrict__ B,
                   const float* __restrict__ bias, const float* __restrict__ residual,
                   float* __restrict__ C, int M, int N, int K, int act)
{
    __shared__ __align__(16) float As[2][AS_ELEMS];  // A tile 64x16 per stage
    __shared__ __align__(16) float Bs[2][BS_ELEMS];  // B tile 16x64 per stage

    const int tid  = threadIdx.x;
    const int wave = tid >> 5;
    const int lid  = tid & 31;
    const int mrow = (wave & 3) * 16;   // wave's row offset in 64-row tile
    const int ncol = (wave >> 2) * 32;  // wave's col offset in 64-col tile
    const int l15  = lid & 15;
    const int hi   = lid >> 4;          // lane group (0: lanes 0-15, 1: 16-31)

    const int rowBase0 = blockIdx.x * 64;   // M blocks on fast grid dim
    const int colBase0 = blockIdx.y * 64;   // N blocks on slow grid dim

    // tile-load assignments (256 threads move 64x16 A + 16x64 B per chunk)
    const int ra = tid >> 2;          // 0..63
    const int ca = (tid & 3) * 4;     // 0,4,8,12
    const int rb = tid >> 4;          // 0..15
    const int cb = (tid & 15) * 4;    // 0..60

    const float* Aload = A + (size_t)(rowBase0 + ra) * K + ca;
    const float* Bload = B + (size_t)rb * N + colBase0 + cb;

    float* Adst0 = &As[0][ra * AS_STRIDE + ca];
    float* Adst1 = &As[1][ra * AS_STRIDE + ca];
    float* Bdst0 = &Bs[0][rb * BS_STRIDE + cb];
    float* Bdst1 = &Bs[1][rb * BS_STRIDE + cb];

    v8f c0 = {};  // cols [ncol, ncol+16)
    v8f c1 = {};  // cols [ncol+16, ncol+32)

    const int nChunks = K >> 4;

#if USE_ASYNC_LDS
    // prologue: issue chunk 0 into stage 0 (2 async B128 copies per thread)
    __builtin_amdgcn_global_load_async_to_lds_b128((gas_v4i)Aload, (lds_v4i)Adst0, 0, 0);
    __builtin_amdgcn_global_load_async_to_lds_b128((gas_v4i)Bload, (lds_v4i)Bdst0, 0, 0);
#else
    float4 a4 = *(const float4*)(Aload);
    float4 b4 = *(const float4*)(Bload);
#endif

    for (int c = 0; c < nChunks; ++c) {
        const int cur = c & 1;
        const int ktn = (c + 1) << 4;
#if USE_ASYNC_LDS
        if (c + 1 < nChunks) {
            float* An = cur ? Adst0 : Adst1;
            float* Bn = cur ? Bdst0 : Bdst1;
            __builtin_amdgcn_global_load_async_to_lds_b128(
                (gas_v4i)(Aload + ktn), (lds_v4i)An, 0, 0);
            __builtin_amdgcn_global_load_async_to_lds_b128(
                (gas_v4i)(Bload + (size_t)ktn * N), (lds_v4i)Bn, 0, 0);
            __builtin_amdgcn_s_wait_asynccnt(2);  // current chunk's copies complete
        } else {
            __builtin_amdgcn_s_wait_asynccnt(0);
        }
#else
        {
            float* Ad = cur ? Adst1 : Adst0;
            float* Bd = cur ? Bdst1 : Bdst0;
            *(float2*)(Ad)     = make_float2(a4.x, a4.y);
            *(float2*)(Ad + 2) = make_float2(a4.z, a4.w);
            *(float2*)(Bd)     = make_float2(b4.x, b4.y);
            *(float2*)(Bd + 2) = make_float2(b4.z, b4.w);
        }
        if (c + 1 < nChunks) {   // issue next chunk's global loads early
            a4 = *(const float4*)(Aload + ktn);
            b4 = *(const float4*)(Bload + (size_t)ktn * N);
            __builtin_prefetch(Aload + ktn + 16, 0, 0);
        }
#endif
        __syncthreads();   // all threads' tile data visible in LDS

        const float* Ac = As[cur];
        const float* Bc = Bs[cur];
#pragma unroll
        for (int kk = 0; kk < 4; ++kk) {
            const int kb = kk * 4 + 2 * hi;   // this lane-group's K pair base
            // A frag 16x4 f32: lane l15 = row M, v0/v1 hold K=kb,kb+1
            const float2 af = *(const float2*)&Ac[(mrow + l15) * AS_STRIDE + kb];
            v2f a = { af.x, af.y };
            // B frag 4x16 f32 (mirrored layout): lane l15 = col N
            v2f b0 = { Bc[kb * BS_STRIDE + ncol + l15],
                       Bc[(kb + 1) * BS_STRIDE + ncol + l15] };
            v2f b1 = { Bc[kb * BS_STRIDE + ncol + 16 + l15],
                       Bc[(kb + 1) * BS_STRIDE + ncol + 16 + l15] };
            c0 = __builtin_amdgcn_wmma_f32_16x16x4_f32(false, a, false, b0, (short)0, c0, false, false);
            c1 = __builtin_amdgcn_wmma_f32_16x16x4_f32(false, a, false, b1, (short)0, c1, false, false);
        }
        __syncthreads();   // protect stage buffer before next overwrite
    }

    // Epilogue: C/D layout -> VGPR j: lanes 0-15 row M=j, lanes 16-31 row M=8+j; N=lane&15
#pragma unroll
    for (int jacc = 0; jacc < 2; ++jacc) {
        const v8f acc = jacc ? c1 : c0;
        const int col = colBase0 + ncol + jacc * 16 + l15;
        const float bv = bias ? bias[col] : 0.0f;
#pragma unroll
        for (int j = 0; j < 8; ++j) {
            const int row = rowBase0 + mrow + j + 8 * hi;
            float val = acc[j] + bv;
            if (act == 1) val = 0.5f * val * (1.0f + erff(val * 0.70710678118654752f)); // exact GELU
            if (residual) val += residual[(size_t)row * N + col];
            C[(size_t)row * N + col] = val;
        }
    }
}

// =====================================================================
// Embedding gather + input LayerNorm: one block per token, 256 threads.
// =====================================================================
__global__ __launch_bounds__(256)
void embed_ln_kernel(const int* __restrict__ ids, const float* __restrict__ embed,
                     const float* __restrict__ g, const float* __restrict__ b,
                     float* __restrict__ out)
{
    const int t = blockIdx.x;
    const int tid = threadIdx.x;
    const float* row = embed + (size_t)ids[t] * D_MODEL;
    float* orow = out + (size_t)t * D_MODEL;

    __shared__ float s1[256], s2[256];
    float v[8];
    float sum = 0.f, sq = 0.f;
#pragma unroll
    for (int i = 0; i < 8; ++i) {
        v[i] = row[tid + i * 256];
        sum += v[i]; sq += v[i] * v[i];
    }
    s1[tid] = sum; s2[tid] = sq;
    __syncthreads();
    for (int off = 128; off > 0; off >>= 1) {
        if (tid < off) { s1[tid] += s1[tid + off]; s2[tid] += s2[tid + off]; }
        __syncthreads();
    }
    const float mean = s1[0] * (1.0f / D_MODEL);
    const float var  = s2[0] * (1.0f / D_MODEL) - mean * mean;
    const float inv  = rsqrtf(var + LN_EPS);
#pragma unroll
    for (int i = 0; i < 8; ++i) {
        const int c = tid + i * 256;
        orow[c] = (v[i] - mean) * inv * g[c] + b[c];
    }
}

// =====================================================================
// LayerNorm over last dim (D_MODEL): one block per token.
// =====================================================================
__global__ __launch_bounds__(256)
void ln_kernel(const float* __restrict__ in, const float* __restrict__ g,
               const float* __restrict__ b, float* __restrict__ out)
{
    const int t = blockIdx.x;
    const int tid = threadIdx.x;
    const float* row = in + (size_t)t * D_MODEL;
    float* orow = out + (size_t)t * D_MODEL;

    __shared__ float s1[256], s2[256];
    float v[8];
    float sum = 0.f, sq = 0.f;
#pragma unroll
    for (int i = 0; i < 8; ++i) {
        v[i] = row[tid + i * 256];
        sum += v[i]; sq += v[i] * v[i];
    }
    s1[tid] = sum; s2[tid] = sq;
    __syncthreads();
    for (int off = 128; off > 0; off >>= 1) {
        if (tid < off) { s1[tid] += s1[tid + off]; s2[tid] += s2[tid + off]; }
        __syncthreads();
    }
    const float mean = s1[0] * (1.0f / D_MODEL);
    const float var  = s2[0] * (1.0f / D_MODEL) - mean * mean;
    const float inv  = rsqrtf(var + LN_EPS);
#pragma unroll
    for (int i = 0; i < 8; ++i) {
        const int c = tid + i * 256;
        orow[c] = (v[i] - mean) * inv * g[c] + b[c];
    }
}

// =====================================================================
// Causal attention, online softmax. One wave32 per (batch, head, query).
// q/k/v/o layout: [TOKENS, D_MODEL] row-major, head h at cols h*128..+127.
// Lane l owns dims 4l..4l+3 (float4); score reduced with __shfl_xor.
// =====================================================================
__global__ __launch_bounds__(256)
void attn_kernel(const float* __restrict__ q, const float* __restrict__ k,
                 const float* __restrict__ v, float* __restrict__ o)
{
    const int wave = threadIdx.x >> 5;
    const int lid  = threadIdx.x & 31;
    const int task = blockIdx.x * 8 + wave;          // 0 .. B*H*S-1
    const int sq = task & (SEQ - 1);
    const int bh = task >> 9;                        // 0..31
    const int h  = bh & (N_HEADS - 1);
    const int b  = bh >> 4;

    const size_t headBase = ((size_t)b * SEQ) * D_MODEL + (size_t)h * HEAD_DIM + lid * 4;
    const size_t qoff = headBase + (size_t)sq * D_MODEL;

    const float4 qv = *(const float4*)(q + qoff);
    float4 oacc = make_float4(0.f, 0.f, 0.f, 0.f);
    float m = -__builtin_inff();
    float l = 0.f;

    for (int sk = 0; sk <= sq; ++sk) {
        const float4 kv = *(const float4*)(k + headBase + (size_t)sk * D_MODEL);
        float p = qv.x * kv.x + qv.y * kv.y + qv.z * kv.z + qv.w * kv.w;
#pragma unroll
        for (int off = 16; off > 0; off >>= 1) p += __shfl_xor(p, off, 32);
        p *= ATT_SCALE;
        const float mn = fmaxf(m, p);
        const float sc = __expf(m - mn);   // 0 on first iter (m = -inf)
        const float e  = __expf(p - mn);
        const float4 vv = *(const float4*)(v + headBase + (size_t)sk * D_MODEL);
        oacc.x = oacc.x * sc + e * vv.x;
        oacc.y = oacc.y * sc + e * vv.y;
        oacc.z = oacc.z * sc + e * vv.z;
        oacc.w = oacc.w * sc + e * vv.w;
        l = l * sc + e;
        m = mn;
    }
    const float inv = 1.0f / l;
    float4 res = make_float4(oacc.x * inv, oacc.y * inv, oacc.z * inv, oacc.w * inv);
    *(float4*)(o + qoff) = res;
}

// =====================================================================
// Host-side orchestration
// =====================================================================
struct LayerPtrs {
    const float *wq, *wk, *wv, *wo;
    const float *ln_attn_g, *ln_attn_b, *ln_ff_g, *ln_ff_b;
    const float *w1, *b1, *w2, *b2;
};

extern "C" void kernel_launch(void* const* d_in, const int* in_sizes, int n_in,
                              void* d_out, int out_size, void* d_ws, size_t ws_size,
                              hipStream_t stream) {
    (void)n_in; (void)out_size; (void)ws_size;

    // ---- resolve input order (insertion order vs jax sorted-key order) ----
    const int* ids = (const int*)d_in[0];
    const float *embed, *ln_in_g, *ln_in_b, *w_out, *b_out;
    LayerPtrs L[4];

    const bool sorted = (in_sizes[1] == VOCAB);   // sorted: b_out comes first
    if (!sorted) {
        int i = 1;
        embed   = (const float*)d_in[i++];
        ln_in_g = (const float*)d_in[i++];
        ln_in_b = (const float*)d_in[i++];
        for (int l = 0; l < 4; ++l) {
            L[l].wq        = (const float*)d_in[i++];
            L[l].wk        = (const float*)d_in[i++];
            L[l].wv        = (const float*)d_in[i++];
            L[l].wo        = (const float*)d_in[i++];
            L[l].ln_attn_g = (const float*)d_in[i++];
            L[l].ln_attn_b = (const float*)d_in[i++];
            L[l].ln_ff_g   = (const float*)d_in[i++];
            L[l].ln_ff_b   = (const float*)d_in[i++];
            L[l].w1        = (const float*)d_in[i++];
            L[l].b1        = (const float*)d_in[i++];
            L[l].w2        = (const float*)d_in[i++];
            L[l].b2        = (const float*)d_in[i++];
        }
        w_out = (const float*)d_in[i++];
        b_out = (const float*)d_in[i++];
    } else {
        // params keys sorted: b_out, embed, layers[...], ln_in_b, ln_in_g, w_out
        // layer keys sorted:  b1,b2,ln_attn_b,ln_attn_g,ln_ff_b,ln_ff_g,w1,w2,wk,wo,wq,wv
        int i = 1;
        b_out = (const float*)d_in[i++];
        embed = (const float*)d_in[i++];
        for (int l = 0; l < 4; ++l) {
            L[l].b1        = (const float*)d_in[i++];
            L[l].b2        = (const float*)d_in[i++];
            L[l].ln_attn_b = (const float*)d_in[i++];
            L[l].ln_attn_g = (const float*)d_in[i++];
            L[l].ln_ff_b   = (const float*)d_in[i++];
            L[l].ln_ff_g   = (const float*)d_in[i++];
            L[l].w1        = (const float*)d_in[i++];
            L[l].w2        = (const float*)d_in[i++];
            L[l].wk        = (const float*)d_in[i++];
            L[l].wo        = (const float*)d_in[i++];
            L[l].wq        = (const float*)d_in[i++];
            L[l].wv        = (const float*)d_in[i++];
        }
        ln_in_b = (const float*)d_in[i++];
        ln_in_g = (const float*)d_in[i++];
        w_out   = (const float*)d_in[i++];
    }

    // ---- workspace partition (floats) ----
    float* ws = (float*)d_ws;
    const size_t TD = (size_t)TOKENS * D_MODEL;       // 2,097,152
    float* x        = ws;                // residual stream
    float* hbuf     = ws + 1 * TD;       // LN output / GEMM input
    float* qb       = ws + 2 * TD;
    float* kb       = ws + 3 * TD;
    float* vb       = ws + 4 * TD;
    float* attn_out = ws + 5 * TD;
    float* ff1      = ws + 6 * TD;       // [TOKENS, FFN] = 4*TD floats
    float* logits   = (float*)d_out;

    auto gemm = [&](const float* A, const float* W, const float* bias,
                    const float* res, float* C, int M, int N, int K, int act) {
        dim3 grid(M / 64, N / 64);   // M-blocks fast -> weight slab L2 reuse
        gemm_f32_wmma<<<grid, 256, 0, stream>>>(A, W, bias, res, C, M, N, K, act);
    };

    // 1) embedding + input LN
    embed_ln_kernel<<<TOKENS, 256, 0, stream>>>(ids, embed, ln_in_g, ln_in_b, x);

    // 2) transformer layers
    for (int l = 0; l < 4; ++l) {
        ln_kernel<<<TOKENS, 256, 0, stream>>>(x, L[l].ln_attn_g, L[l].ln_attn_b, hbuf);
        gemm(hbuf, L[l].wq, nullptr, nullptr, qb, TOKENS, D_MODEL, D_MODEL, 0);
        gemm(hbuf, L[l].wk, nullptr, nullptr, kb, TOKENS, D_MODEL, D_MODEL, 0);
        gemm(hbuf, L[l].wv, nullptr, nullptr, vb, TOKENS, D_MODEL, D_MODEL, 0);
        attn_kernel<<<(BATCH * N_HEADS * SEQ) / 8, 256, 0, stream>>>(qb, kb, vb, attn_out);
        gemm(attn_out, L[l].wo, nullptr, x, x, TOKENS, D_MODEL, D_MODEL, 0);  // x += o @ wo
        ln_kernel<<<TOKENS, 256, 0, stream>>>(x, L[l].ln_ff_g, L[l].ln_ff_b, hbuf);
        gemm(hbuf, L[l].w1, L[l].b1, nullptr, ff1, TOKENS, FFN, D_MODEL, 1);  // exact GELU
        gemm(ff1, L[l].w2, L[l].b2, x, x, TOKENS, D_MODEL, FFN, 0);           // x += ffn
    }

    // 3) logits
    gemm(x, w_out, b_out, nullptr, logits, TOKENS, VOCAB, D_MODEL, 0);
}